// Model_65773129171355
// MI455X (gfx1250) — compile-verified
//
#include <hip/hip_runtime.h>
#include <math.h>

typedef __attribute__((ext_vector_type(2))) float v2f;
typedef __attribute__((ext_vector_type(8))) float v8f;

#define T_N        64
#define SEGS       64
#define PTS        65
#define H_DIM      128
#define W_DIM      128
#define W_SHIFT    7          // log2(W_DIM)
#define SEG_TILES  (SEGS / 16)
#define PIX        (H_DIM * W_DIM)
#define PIX_TILES  (PIX / 16)
#define NWAVES     4
#define BLOCK      (NWAVES * 32)
#define NSPLIT     8          // blocks per worm (pixel-tile split) for occupancy

// delta_max[t,h,w] = max_s( wid[s] - sqrt((mx[s]-w)^2 + (my[s]-h)^2) )
// d^2 via V_WMMA_F32_16X16X4_F32 as A[16x4] x B[4x16]:
//   A[s,:] = { mx^2+my^2, 1, -2mx, -2my }   (per-worm, registers)
//   B[:,p] = { 1, wx^2+hy^2, wx, hy }       (per-pixel-tile)
__global__ __launch_bounds__(BLOCK) void worm_render_kernel(
    const float* __restrict__ cx, const float* __restrict__ cy,
    const float* __restrict__ theta, const float* __restrict__ uL,
    const float* __restrict__ alpha_p, const float* __restrict__ gamma_p,
    const float* __restrict__ delta_p, float* __restrict__ out)
{
    __shared__ float sh_c[SEGS], sh_s[SEGS];
    __shared__ float sh_x[PTS], sh_y[PTS];
    __shared__ float sh_Am[SEGS], sh_Ax[SEGS], sh_Ay[SEGS], sh_wid[SEGS];
    __shared__ float sh_mean[2];

    const int t   = blockIdx.x;
    const int tid = threadIdx.x;

    const float alpha = alpha_p[0];
    const float gamma = gamma_p[0];
    const float delta = delta_p[0];

    // ---- Setup (precise math; once per block) ----
    if (tid < SEGS) {
        float th = theta[t * SEGS + tid];
        float u  = uL[t];
        sh_c[tid] = u * cosf(th);
        sh_s[tid] = u * sinf(th);

        float ds  = 1.0f / (1.0f + expf(-delta));
        float ge  = 0.5f + expf(gamma);
        float wxl = -1.0f + 2.0f * (float)tid / (float)(SEGS - 1);
        float ax  = fabsf(wxl);
        float pw  = powf(ax, 2.0f * ge);
        sh_wid[tid] = alpha * sqrtf(1.0f + 1e-5f -
                                    pw * (1.0f + 2.0f * ge * ds - 2.0f * ge * ds * ax));
    }
    __syncthreads();

    if (tid == 0) {
        float run = 0.0f, sum = 0.0f;
        sh_x[0] = 0.0f;
        for (int i = 0; i < SEGS; ++i) { run += sh_c[i]; sh_x[i + 1] = run; sum += run; }
        sh_mean[0] = sum / (float)PTS;
        run = 0.0f; sum = 0.0f;
        sh_y[0] = 0.0f;
        for (int i = 0; i < SEGS; ++i) { run += sh_s[i]; sh_y[i + 1] = run; sum += run; }
        sh_mean[1] = sum / (float)PTS;
    }
    __syncthreads();

    if (tid < SEGS) {
        float mx = 0.5f * (sh_x[tid] + sh_x[tid + 1]) - sh_mean[0] + cx[t];
        float my = 0.5f * (sh_y[tid] + sh_y[tid + 1]) - sh_mean[1] + cy[t];
        sh_Am[tid] = mx * mx + my * my;
        sh_Ax[tid] = -2.0f * mx;
        sh_Ay[tid] = -2.0f * my;
    }
    __syncthreads();

    const unsigned lane = tid & 31u;
    const unsigned wave = (unsigned)tid >> 5;
    const unsigned half = lane >> 4;   // 0: lanes 0-15 (K=0,1), 1: lanes 16-31 (K=2,3)
    const unsigned nidx = lane & 15u;

    // A tiles + per-row widths in registers, reused across all pixel tiles.
    v2f   A[SEG_TILES];
    float Wd[SEG_TILES][8];
#pragma unroll
    for (int st = 0; st < SEG_TILES; ++st) {
        int s = st * 16 + (int)nidx;
        A[st][0] = half ? sh_Ax[s] : sh_Am[s];
        A[st][1] = half ? sh_Ay[s] : 1.0f;
#pragma unroll
        for (int r = 0; r < 8; ++r)
            Wd[st][r] = sh_wid[st * 16 + r + 8 * (int)half];   // C/D row = r + 8*half
    }

    // ---- Main loop over 16-pixel tiles (h fixed per tile; W % 16 == 0) ----
    for (unsigned pt = blockIdx.y * NWAVES + wave; pt < PIX_TILES;
         pt += NWAVES * NSPLIT) {
        unsigned p0 = pt << 4;
        unsigned h  = p0 >> W_SHIFT;          // unsigned: pure shift
        unsigned w0 = p0 & (W_DIM - 1u);      // unsigned: pure mask

        float wx = (float)(w0 + nidx);
        float hy = (float)h;

        // B (4x16 f32), branchless half-wave select
        v2f B;
        B[0] = half ? wx : 1.0f;
        B[1] = half ? hy : fmaf(wx, wx, hy * hy);

        // Issue all 4 WMMAs first (distinct accumulators -> back-to-back matrix ops)
        v8f acc[SEG_TILES];
#pragma unroll
        for (int st = 0; st < SEG_TILES; ++st) {
            v8f c = {0.f, 0.f, 0.f, 0.f, 0.f, 0.f, 0.f, 0.f};
            acc[st] = __builtin_amdgcn_wmma_f32_16x16x4_f32(
                          false, A[st], false, B, (short)0, c, false, false);
        }

        float best = -3.0e38f;
#pragma unroll
        for (int st = 0; st < SEG_TILES; ++st) {
#pragma unroll
            for (int r = 0; r < 8; ++r) {
                // |.| folds into v_sqrt_f32 src modifier; guards rounding-negative d2
                float dist = __builtin_amdgcn_sqrtf(fabsf(acc[st][r]));
                best = fmaxf(best, Wd[st][r] - dist);
            }
        }

        // combine lane halves: lane l holds rows {0..7}+8*half for column N=l&15
        best = fmaxf(best, __shfl_xor(best, 16, 32));

        if (lane < 16) {
            float sig = 1.0f / (1.0f + __expf(-2.0f * best));   // SHARPNESS = 2
            out[t * PIX + (int)(h * W_DIM + w0 + nidx)] =
                255.0f * (1.2f * (sig - 0.5f) + 0.5f);          // CONTRAST = 1.2
        }
    }
}

extern "C" void kernel_launch(void* const* d_in, const int* in_sizes, int n_in,
                              void* d_out, int out_size, void* d_ws, size_t ws_size,
                              hipStream_t stream) {
    (void)in_sizes; (void)n_in; (void)out_size; (void)d_ws; (void)ws_size;
    const float* cx    = (const float*)d_in[0];
    const float* cy    = (const float*)d_in[1];
    const float* theta = (const float*)d_in[2];
    const float* uL    = (const float*)d_in[3];
    const float* alpha = (const float*)d_in[4];
    const float* gamma = (const float*)d_in[5];
    const float* delta = (const float*)d_in[6];
    float* out = (float*)d_out;

    worm_render_kernel<<<dim3(T_N, NSPLIT), dim3(BLOCK), 0, stream>>>(
        cx, cy, theta, uL, alpha, gamma, delta, out);
}